// HypergraphLayer_53283364274773
// MI455X (gfx1250) — compile-verified
//
#include <hip/hip_runtime.h>

// Problem constants (reference: T,N,E,D,NL = 2,4096,4096,128,2)
#define TT 2
#define NN 4096
#define EE 4096
#define DD 128
#define NLAY 2
#define CAP (1 << 18)   // sparse index capacity (expected total nnz ~67K; 4x headroom)

typedef __attribute__((ext_vector_type(2))) float v2f;
typedef __attribute__((ext_vector_type(8))) float v8f;

// ------------------------------------------------------------------
// init: node accumulator = x, review accumulator = 0, counters = 0
// ------------------------------------------------------------------
__global__ void hg_init(const float* __restrict__ x, float* __restrict__ nacc,
                        float* __restrict__ rev, int* __restrict__ colcnt,
                        int* __restrict__ cursor) {
  int stride = gridDim.x * blockDim.x;
  for (int i = blockIdx.x * blockDim.x + threadIdx.x; i < NN * DD; i += stride) {
    nacc[i] = x[i];
    if (i < EE * DD) rev[i] = 0.0f;
    if (i < TT * EE) { colcnt[i] = 0; cursor[i] = 0; }
  }
}

// ------------------------------------------------------------------
// one wave per H row (t,n): row nnz -> rowcnt/dVis, col counts via int atomics
// ------------------------------------------------------------------
__global__ void hg_degrees(const float* __restrict__ H, int* __restrict__ rowcnt,
                           int* __restrict__ colcnt, float* __restrict__ dVis) {
  int lane = threadIdx.x & 31;
  int row = (blockIdx.x * blockDim.x + threadIdx.x) >> 5;
  if (row >= TT * NN) return;
  int t = row / NN;
  const float* hrow = H + (size_t)row * EE;
  int total = 0;
  for (int base = 0; base < EE; base += 32) {
    bool nz = hrow[base + lane] != 0.0f;
    unsigned m = (unsigned)__ballot(nz);
    if (nz) atomicAdd(&colcnt[t * EE + base + lane], 1);
    total += __popc(m);
  }
  if (lane == 0) {
    rowcnt[row] = total;
    dVis[row] = (total > 0) ? (1.0f / sqrtf((float)total)) : 0.0f;
  }
}

// ------------------------------------------------------------------
// single-block exclusive scan (n <= a few * 1024 chunks), writes total at out[n]
// ------------------------------------------------------------------
__global__ void hg_scan_excl(const int* __restrict__ in, int* __restrict__ out, int n) {
  __shared__ int sh[1024];
  __shared__ int carry_s;
  if (threadIdx.x == 0) carry_s = 0;
  __syncthreads();
  for (int base = 0; base < n; base += 1024) {
    int i = base + (int)threadIdx.x;
    int v = (i < n) ? in[i] : 0;
    sh[threadIdx.x] = v;
    __syncthreads();
    for (int off = 1; off < 1024; off <<= 1) {
      int add = (threadIdx.x >= (unsigned)off) ? sh[threadIdx.x - off] : 0;
      __syncthreads();
      sh[threadIdx.x] += add;
      __syncthreads();
    }
    int carry = carry_s;
    if (i < n) out[i] = carry + sh[threadIdx.x] - v;  // exclusive
    __syncthreads();
    if (threadIdx.x == 1023) carry_s = carry + sh[1023];
    __syncthreads();
  }
  if (threadIdx.x == 0) out[n] = carry_s;
}

__global__ void hg_einv(const int* __restrict__ colcnt, float* __restrict__ dEinv) {
  int i = blockIdx.x * blockDim.x + threadIdx.x;
  if (i < TT * EE) {
    int c = colcnt[i];
    dEinv[i] = (c > 0) ? (1.0f / (float)c) : 0.0f;
  }
}

// ------------------------------------------------------------------
// CSR fill: wave per row, ballot-compaction keeps edge ids ascending (deterministic)
// ------------------------------------------------------------------
__global__ void hg_csr_fill(const float* __restrict__ H, const int* __restrict__ csr_ptr,
                            int* __restrict__ csr_idx) {
  int lane = threadIdx.x & 31;
  int row = (blockIdx.x * blockDim.x + threadIdx.x) >> 5;
  if (row >= TT * NN) return;
  const float* hrow = H + (size_t)row * EE;
  int cursor = csr_ptr[row];
  for (int base = 0; base < EE; base += 32) {
    bool nz = hrow[base + lane] != 0.0f;
    unsigned m = (unsigned)__ballot(nz);
    if (nz) {
      int pos = cursor + __popc(m & ((1u << lane) - 1u));
      if (pos < CAP) csr_idx[pos] = base + lane;
    }
    cursor += __popc(m);
  }
}

// CSC fill via atomic cursors (order fixed afterwards by per-edge sort)
__global__ void hg_csc_fill(const float* __restrict__ H, const int* __restrict__ csc_ptr,
                            int* __restrict__ cursor, int* __restrict__ csc_idx) {
  int lane = threadIdx.x & 31;
  int row = (blockIdx.x * blockDim.x + threadIdx.x) >> 5;
  if (row >= TT * NN) return;
  int t = row / NN;
  int n = row - t * NN;
  const float* hrow = H + (size_t)row * EE;
  for (int base = 0; base < EE; base += 32) {
    int e = base + lane;
    if (hrow[e] != 0.0f) {
      int seg = t * EE + e;
      int p = atomicAdd(&cursor[seg], 1);
      int pos = csc_ptr[seg] + p;
      if (pos < CAP) csc_idx[pos] = n;
    }
  }
}

// sort each edge's node list ascending (deg ~ 8) -> fully deterministic structure
__global__ void hg_csc_sort(const int* __restrict__ csc_ptr, int* __restrict__ csc_idx) {
  int s = blockIdx.x * blockDim.x + threadIdx.x;
  if (s >= TT * EE) return;
  int beg = csc_ptr[s];
  int end = csc_ptr[s + 1]; if (end > CAP) end = CAP;
  for (int i = beg + 1; i < end; ++i) {
    int key = csc_idx[i];
    int j = i - 1;
    while (j >= beg && csc_idx[j] > key) { csc_idx[j + 1] = csc_idx[j]; --j; }
    csc_idx[j + 1] = key;
  }
}

// ------------------------------------------------------------------
// WMMA f32 GEMM: Y[t,n,o] = dVis[t,n] * (X[n,:] . Wl[t,o,:] + bl[t,o])
// one wave per 16x16 output tile; K accumulated 4 at a time via
// V_WMMA_F32_16X16X4_F32 (full fp32 precision)
// ------------------------------------------------------------------
__global__ void hg_gemm_y(const float* __restrict__ X, const float* __restrict__ Wl,
                          const float* __restrict__ bl, const float* __restrict__ dVis,
                          float* __restrict__ Y) {
  int lane = threadIdx.x & 31;
  int wave = threadIdx.x >> 5;
  int tile = blockIdx.x * (blockDim.x >> 5) + wave;   // [0, TT*(NN/16)*(DD/16))
  const int tilesPerType = (NN / 16) * (DD / 16);     // 2048
  int t = tile / tilesPerType;
  int rem = tile - t * tilesPerType;
  int n0 = (rem >> 3) << 4;          // node tile base
  int o0 = (rem & 7) << 4;           // output-feature tile base
  int r = lane & 15;                 // row/col within tile
  int h = lane >> 4;                 // half select (K split)
  // A frag (16x4 f32): lane holds X[n0+r, k0+2h .. k0+2h+1]
  const float* arow = X + (size_t)(n0 + r) * DD + 2 * h;
  // B frag (4x16 f32): lane holds W^T[k0+2h .. +1, o0+r] = Wl[t, o0+r, k0+2h ..]
  const float* brow = Wl + (size_t)t * DD * DD + (size_t)(o0 + r) * DD + 2 * h;
  v8f c = {0.f, 0.f, 0.f, 0.f, 0.f, 0.f, 0.f, 0.f};
#pragma unroll
  for (int k0 = 0; k0 < DD; k0 += 4) {
    v2f a = *(const v2f*)(arow + k0);
    v2f b = *(const v2f*)(brow + k0);
    c = __builtin_amdgcn_wmma_f32_16x16x4_f32(false, a, false, b, (short)0, c,
                                              false, false);
  }
  float bias = bl[t * DD + o0 + r];
#pragma unroll
  for (int v = 0; v < 8; ++v) {
    int rowN = n0 + v + 8 * h;       // C/D layout: VGPR v -> M = v (+8 for hi lanes)
    float s = dVis[t * NN + rowN];
    Y[(size_t)t * NN * DD + (size_t)rowN * DD + (o0 + r)] = s * (c[v] + bias);
  }
}

// ------------------------------------------------------------------
// ye[t,e,:] = dEinv[t,e] * sum_{n in edge e} Y[t,n,:]   (wave per (t,e), float4/lane)
// ------------------------------------------------------------------
__global__ void hg_edge_gather(const float* __restrict__ Y, const int* __restrict__ csc_ptr,
                               const int* __restrict__ csc_idx, const float* __restrict__ dEinv,
                               float* __restrict__ YE) {
  int lane = threadIdx.x & 31;
  int te = (blockIdx.x * blockDim.x + threadIdx.x) >> 5;
  if (te >= TT * EE) return;
  int t = te / EE;
  int beg = csc_ptr[te];
  int end = csc_ptr[te + 1]; if (end > CAP) end = CAP;
  const float* Yt = Y + (size_t)t * NN * DD;
  float a0 = 0.f, a1 = 0.f, a2 = 0.f, a3 = 0.f;
  for (int i = beg; i < end; ++i) {
    int n = csc_idx[i];
    if (i + 1 < end)
      __builtin_prefetch(Yt + (size_t)csc_idx[i + 1] * DD + lane * 4, 0, 1);
    float4 v = *(const float4*)(Yt + (size_t)n * DD + lane * 4);
    a0 += v.x; a1 += v.y; a2 += v.z; a3 += v.w;
  }
  float s = dEinv[te];
  float4 o; o.x = s * a0; o.y = s * a1; o.z = s * a2; o.w = s * a3;
  *(float4*)(YE + (size_t)te * DD + lane * 4) = o;
}

// ------------------------------------------------------------------
// e[t,n,:] = leaky( dVis[t,n] * sum_{e in node n} YE[t,e,:], 0.01 )
// ------------------------------------------------------------------
__global__ void hg_node_gather(const float* __restrict__ YE, const int* __restrict__ csr_ptr,
                               const int* __restrict__ csr_idx, const float* __restrict__ dVis,
                               float* __restrict__ EB) {
  int lane = threadIdx.x & 31;
  int tn = (blockIdx.x * blockDim.x + threadIdx.x) >> 5;
  if (tn >= TT * NN) return;
  int t = tn / NN;
  int beg = csr_ptr[tn];
  int end = csr_ptr[tn + 1]; if (end > CAP) end = CAP;
  const float* YEt = YE + (size_t)t * EE * DD;
  float a0 = 0.f, a1 = 0.f, a2 = 0.f, a3 = 0.f;
  for (int i = beg; i < end; ++i) {
    int e = csr_idx[i];
    if (i + 1 < end)
      __builtin_prefetch(YEt + (size_t)csr_idx[i + 1] * DD + lane * 4, 0, 1);
    float4 v = *(const float4*)(YEt + (size_t)e * DD + lane * 4);
    a0 += v.x; a1 += v.y; a2 += v.z; a3 += v.w;
  }
  float s = dVis[tn];
  float4 o;
  o.x = s * a0; o.x = (o.x > 0.f) ? o.x : 0.01f * o.x;
  o.y = s * a1; o.y = (o.y > 0.f) ? o.y : 0.01f * o.y;
  o.z = s * a2; o.z = (o.z > 0.f) ? o.z : 0.01f * o.z;
  o.w = s * a3; o.w = (o.w > 0.f) ? o.w : 0.01f * o.w;
  *(float4*)(EB + (size_t)tn * DD + lane * 4) = o;
}

// ------------------------------------------------------------------
// rev[e,:] += sum_t dEinv[t,e] * sum_{n in edge e} EB[t,n,:]   (wave per e)
// ------------------------------------------------------------------
__global__ void hg_review(const float* __restrict__ EB, const int* __restrict__ csc_ptr,
                          const int* __restrict__ csc_idx, const float* __restrict__ dEinv,
                          float* __restrict__ rev) {
  int lane = threadIdx.x & 31;
  int e = (blockIdx.x * blockDim.x + threadIdx.x) >> 5;
  if (e >= EE) return;
  float a0 = 0.f, a1 = 0.f, a2 = 0.f, a3 = 0.f;
  for (int t = 0; t < TT; ++t) {
    int seg = t * EE + e;
    int beg = csc_ptr[seg];
    int end = csc_ptr[seg + 1]; if (end > CAP) end = CAP;
    float s = dEinv[seg];
    const float* EBt = EB + (size_t)t * NN * DD;
    float b0 = 0.f, b1 = 0.f, b2 = 0.f, b3 = 0.f;
    for (int i = beg; i < end; ++i) {
      int n = csc_idx[i];
      if (i + 1 < end)
        __builtin_prefetch(EBt + (size_t)csc_idx[i + 1] * DD + lane * 4, 0, 1);
      float4 v = *(const float4*)(EBt + (size_t)n * DD + lane * 4);
      b0 += v.x; b1 += v.y; b2 += v.z; b3 += v.w;
    }
    a0 += s * b0; a1 += s * b1; a2 += s * b2; a3 += s * b3;
  }
  float* o = rev + (size_t)e * DD + lane * 4;
  o[0] += a0; o[1] += a1; o[2] += a2; o[3] += a3;
}

// x_next = EB[0] + EB[1]; node accumulator += x_next
__global__ void hg_xnext(const float* __restrict__ EB, float* __restrict__ XN,
                         float* __restrict__ nacc) {
  int stride = gridDim.x * blockDim.x;
  for (int i = blockIdx.x * blockDim.x + threadIdx.x; i < NN * DD; i += stride) {
    float v = EB[i] + EB[(size_t)NN * DD + i];
    XN[i] = v;
    nacc[i] += v;
  }
}

__global__ void hg_finalize(const float* __restrict__ nacc, const float* __restrict__ rev,
                            float* __restrict__ out) {
  int stride = gridDim.x * blockDim.x;
  for (int i = blockIdx.x * blockDim.x + threadIdx.x; i < NN * DD; i += stride)
    out[i] = nacc[i] * (1.0f / (float)(NLAY + 1));
  for (int i = blockIdx.x * blockDim.x + threadIdx.x; i < EE * DD; i += stride)
    out[(size_t)NN * DD + i] = rev[i] * (1.0f / (float)NLAY);
}

// ------------------------------------------------------------------
extern "C" void kernel_launch(void* const* d_in, const int* in_sizes, int n_in,
                              void* d_out, int out_size, void* d_ws, size_t ws_size,
                              hipStream_t stream) {
  (void)in_sizes; (void)n_in; (void)out_size; (void)ws_size;
  const float* x = (const float*)d_in[0];   // [N,D]
  const float* H = (const float*)d_in[1];   // [T,N,E]
  const float* W = (const float*)d_in[2];   // [NL,T,D,D]
  const float* b = (const float*)d_in[3];   // [NL,T,D]

  char* base = (char*)d_ws;
  size_t off = 0;
  auto take = [&](size_t bytes) -> void* {
    void* p = (void*)(base + off);
    off += (bytes + 255) & ~(size_t)255;
    return p;
  };
  float* dVis    = (float*)take((size_t)TT * NN * 4);
  float* dEinv   = (float*)take((size_t)TT * EE * 4);
  int*   rowcnt  = (int*)take((size_t)TT * NN * 4);
  int*   colcnt  = (int*)take((size_t)TT * EE * 4);
  int*   cursor  = (int*)take((size_t)TT * EE * 4);
  int*   csr_ptr = (int*)take(((size_t)TT * NN + 1) * 4);
  int*   csc_ptr = (int*)take(((size_t)TT * EE + 1) * 4);
  int*   csr_idx = (int*)take((size_t)CAP * 4);
  int*   csc_idx = (int*)take((size_t)CAP * 4);
  float* Y       = (float*)take((size_t)TT * NN * DD * 4);
  float* YE      = (float*)take((size_t)TT * EE * DD * 4);
  float* EB      = (float*)take((size_t)TT * NN * DD * 4);
  float* XN      = (float*)take((size_t)NN * DD * 4);
  float* NACC    = (float*)take((size_t)NN * DD * 4);
  float* REV     = (float*)take((size_t)EE * DD * 4);

  // --- structural pass: three scans of dense H (134 MB each) ---
  hg_init<<<512, 256, 0, stream>>>(x, NACC, REV, colcnt, cursor);
  hg_degrees<<<1024, 256, 0, stream>>>(H, rowcnt, colcnt, dVis);
  hg_scan_excl<<<1, 1024, 0, stream>>>(rowcnt, csr_ptr, TT * NN);
  hg_scan_excl<<<1, 1024, 0, stream>>>(colcnt, csc_ptr, TT * EE);
  hg_einv<<<32, 256, 0, stream>>>(colcnt, dEinv);
  hg_csr_fill<<<1024, 256, 0, stream>>>(H, csr_ptr, csr_idx);
  hg_csc_fill<<<1024, 256, 0, stream>>>(H, csc_ptr, cursor, csc_idx);
  hg_csc_sort<<<32, 256, 0, stream>>>(csc_ptr, csc_idx);

  // --- layers ---
  for (int l = 0; l < NLAY; ++l) {
    const float* Xsrc = (l == 0) ? x : XN;
    const float* Wl = W + (size_t)l * TT * DD * DD;
    const float* bl = b + (size_t)l * TT * DD;
    hg_gemm_y<<<512, 256, 0, stream>>>(Xsrc, Wl, bl, dVis, Y);          // WMMA
    hg_edge_gather<<<1024, 256, 0, stream>>>(Y, csc_ptr, csc_idx, dEinv, YE);
    hg_node_gather<<<1024, 256, 0, stream>>>(YE, csr_ptr, csr_idx, dVis, EB);
    hg_review<<<512, 256, 0, stream>>>(EB, csc_ptr, csc_idx, dEinv, REV);
    hg_xnext<<<512, 256, 0, stream>>>(EB, XN, NACC);
  }

  hg_finalize<<<1024, 256, 0, stream>>>(NACC, REV, (float*)d_out);
}